// HighPassBiquad_62208306315641
// MI455X (gfx1250) — compile-verified
//
#include <hip/hip_runtime.h>

typedef __attribute__((ext_vector_type(2))) float v2f;
typedef __attribute__((ext_vector_type(8))) float v8f;

#define TAPS        128
#define P_TERMS     6            // taps covered: k in [0, 80+i]; g[81] ~ 2e-10 << f32 eps
#define WG_THREADS  256          // 8 wave32s
#define OUT_PER_WAVE 256         // 16x16 WMMA tile of outputs per wave
#define OUT_PER_WG  2048
#define HEAD        128          // history for taps + reflect pad region
#define LDS_DATA    (HEAD + OUT_PER_WG)   // 2176 floats (unswizzled index space)
#define LDS_DATA_SW (LDS_DATA + (LDS_DATA / 64) * 4)   // +16B pad per 64 floats
#define GREV_LEN    144          // reversed g + 16-entry zero tail for k<0 taps
#define PAD         100

// pad-swizzle: 4-float pad every 64 floats -> conflict-free 64B-stride lane reads
__device__ __forceinline__ int swz(int idx) { return idx + ((idx >> 6) << 2); }

__global__ __launch_bounds__(WG_THREADS)
void biquad_wmma_kernel(const float* __restrict__ x, float* __restrict__ y, int N)
{
    __shared__ float s_data[LDS_DATA_SW];
    __shared__ float s_grev[GREV_LEN];

    const int tid = threadIdx.x;
    const long long n0 = (long long)blockIdx.x * OUT_PER_WG;

    // ---- thread 0: biquad impulse response, stored REVERSED with zero tail ----
    // s_grev[t] = g[127 - t] for t<=127, 0 for t in [128,144)  (covers k<0 taps)
    if (tid == 0) {
        const double cw = 0.9238795325112867;   // cos(pi/8)
        const double sw = 0.3826834323650898;   // sin(pi/8)
        const double Qv = 0.7071067811865476;
        double alpha = sw / (2.0 * Qv);
        double a0 = 1.0 + alpha;
        double b0 = (1.0 + cw) * 0.5 / a0;
        double b1 = -(1.0 + cw) / a0;
        double b2 = b0;
        double a1 = (-2.0 * cw) / a0;
        double a2 = (1.0 - alpha) / a0;
        double y1 = 0.0, y2 = 0.0;
        for (int k = 0; k < TAPS; ++k) {
            double z = (k == 0) ? b0 : (k == 1) ? b1 : (k == 2) ? b2 : 0.0;
            double yk = z - a1 * y1 - a2 * y2;
            s_grev[TAPS - 1 - k] = (float)yk;
            y2 = y1; y1 = yk;
        }
        for (int t = TAPS; t < GREV_LEN; ++t) s_grev[t] = 0.0f;
    }

    // ---- stage padded signal s[m] into LDS (swizzled) ----
    // s[m] = x[m] for m>=0 (tail-clamped values feed only discarded outputs),
    // reflected x[-1-m] for -PAD<=m<0, 0 for m<-PAD (zero-state scan history).
    auto load_s = [&](long long m) -> float {
        if (m >= 0) {
            long long mm = (m < (long long)N) ? m : (long long)N - 1;
            return x[mm];
        }
        if (m >= -(long long)PAD) return x[-1 - m];
        return 0.0f;
    };

    const bool interior = (blockIdx.x > 0) && (n0 + OUT_PER_WG <= (long long)N);
    if (interior) {
        // bulk DMA: GLOBAL_LOAD_ASYNC_TO_LDS_B128 (ASYNCcnt), 544 x 16B chunks.
        // Each b128 (floats fidx..fidx+3, fidx%4==0) stays contiguous under swz().
        for (int q = tid; q < LDS_DATA / 4; q += WG_THREADS) {
            int fidx = 4 * q;
            const float* gp = x + (n0 - HEAD + fidx);
            unsigned lds_byte = (unsigned)(uintptr_t)(&s_data[swz(fidx)]);
            asm volatile("global_load_async_to_lds_b128 %0, %1, off"
                         :: "v"(lds_byte), "v"(gp) : "memory");
        }
    } else {
        // edge blocks: scalar path handles reflect pad / tail clamp
        for (int fidx = tid; fidx < LDS_DATA; fidx += WG_THREADS) {
            s_data[swz(fidx)] = load_s(n0 - HEAD + fidx);
        }
    }
    asm volatile("s_wait_asynccnt 0x0" ::: "memory");
    __syncthreads();

    const int wave  = tid >> 5;       // wave32
    const int lane  = tid & 31;
    const int i_row = lane & 15;      // A row i / B,D column j
    const int kh    = lane >> 4;      // K half select

    // ---- A operands: block-Toeplitz coefficient chunks, ascending-contiguous ----
    // lane needs {g[k0], g[k0-1]} = {s_grev[127-k0], s_grev[128-k0]}; zero tail
    // makes all k<0 boundary cases branch-free. t in [32, 142].
    v2f a_reg[P_TERMS][4];
#pragma unroll
    for (int p = 0; p < P_TERMS; ++p) {
#pragma unroll
        for (int c = 0; c < 4; ++c) {
            int k0 = 16 * p + i_row - (4 * c + 2 * kh);
            int t  = (TAPS - 1) - k0;
            v2f a;
            a.x = s_grev[t];
            a.y = s_grev[t + 1];
            a_reg[p][c] = a;
        }
    }

    // ---- Y = sum_p G_p x S_p via 24 x v_wmma_f32_16x16x4_f32 ----
    const int wbase = HEAD + wave * OUT_PER_WAVE;   // unswizzled LDS index of s[n0w]
    v8f acc = {};
#pragma unroll
    for (int p = 0; p < P_TERMS; ++p) {
#pragma unroll
        for (int c = 0; c < 4; ++c) {
            // B_p[r][j] = s[n0w + 16(j-p) + r], lane: j=i_row, r=4c+2kh (+1)
            int idx = wbase + 16 * (i_row - p) + 4 * c + 2 * kh;   // even
            v2f b = *(const v2f*)(&s_data[swz(idx)]);   // pair stays in one group
            acc = __builtin_amdgcn_wmma_f32_16x16x4_f32(
                /*neg_a=*/false, a_reg[p][c],
                /*neg_b=*/false, b,
                /*c_mod=*/(short)0, acc,
                /*reuse_a=*/false, /*reuse_b=*/false);
        }
    }

    // ---- store: lane holds 8 consecutive outputs y[base .. base+7] ----
    long long base = n0 + (long long)wave * OUT_PER_WAVE + 16 * i_row + 8 * kh;
    if (base + 7 < (long long)N) {
        float4 lo = make_float4(acc[0], acc[1], acc[2], acc[3]);
        float4 hi = make_float4(acc[4], acc[5], acc[6], acc[7]);
        *(float4*)(&y[base])     = lo;
        *(float4*)(&y[base + 4]) = hi;
    } else {
        float tmp[8] = {acc[0], acc[1], acc[2], acc[3], acc[4], acc[5], acc[6], acc[7]};
        for (int v = 0; v < 8; ++v) {
            long long o = base + v;
            if (o < (long long)N) y[o] = tmp[v];
        }
    }
}

extern "C" void kernel_launch(void* const* d_in, const int* in_sizes, int n_in,
                              void* d_out, int out_size, void* d_ws, size_t ws_size,
                              hipStream_t stream) {
    const float* x = (const float*)d_in[0];
    float* yout = (float*)d_out;
    int N = in_sizes[0];
    int grid = (N + OUT_PER_WG - 1) / OUT_PER_WG;
    hipLaunchKernelGGL(biquad_wmma_kernel, dim3(grid), dim3(WG_THREADS), 0, stream,
                       x, yout, N);
}